// SKL_36567351558515
// MI455X (gfx1250) — compile-verified
//
#include <hip/hip_runtime.h>
#include <hip/hip_bf16.h>
#include <math.h>

// Problem constants (reference: B=256, D=4096, K=32)
#define BROW 256
#define DDIM 4096
#define KEIG 32
#define NSVD 288   // K + B

typedef __attribute__((ext_vector_type(2))) float v2f;
typedef __attribute__((ext_vector_type(8))) float v8f;

// ---------------------------------------------------------------------------
// WMMA f32 16x16x4 helper: one wave computes one 16x16 f32 tile.
//   A-layout (16x4):  lane m=lane&15 holds A[m][kb], A[m][kb+1], kb=(lane>>4)*2
//   B-layout (4x16):  lane n=lane&15 holds B[kb][n], B[kb+1][n]
//   C/D (16x16, v8f): vgpr r: lanes 0-15 -> M=r, lanes 16-31 -> M=r+8
// ---------------------------------------------------------------------------
__device__ __forceinline__ v8f wmma_f32(v2f a, v2f b, v8f c) {
  return __builtin_amdgcn_wmma_f32_16x16x4_f32(false, a, false, b, (short)0, c,
                                               false, false);
}

// ---------------------------------------------------------------------------
// 1) new_mean = 0.9*mean + 0.1*colmean(inputs);  centered = inputs - new_mean
// ---------------------------------------------------------------------------
__global__ void mean_center_kernel(const float* __restrict__ X,
                                   const float* __restrict__ mean,
                                   float* __restrict__ nm_out,
                                   float* __restrict__ centered) {
  int d = blockIdx.x * blockDim.x + threadIdx.x;
  if (d >= DDIM) return;
  float s = 0.f;
  for (int i = 0; i < BROW; ++i) s += X[i * DDIM + d];
  float nm = 0.9f * mean[d] + 0.1f * (s * (1.0f / (float)BROW));
  nm_out[d] = nm;
  for (int i = 0; i < BROW; ++i) centered[i * DDIM + d] = X[i * DDIM + d] - nm;
}

// ---------------------------------------------------------------------------
// 2) P[256x32] = centered[256x4096] @ U[4096x32]      (WMMA, 32 tiles)
// ---------------------------------------------------------------------------
__global__ __launch_bounds__(256) void gemm_P_kernel(
    const float* __restrict__ C, const float* __restrict__ U,
    float* __restrict__ P) {
  int wave = (blockIdx.x * blockDim.x + threadIdx.x) >> 5;
  int lane = threadIdx.x & 31;
  int ti = wave >> 1, tj = wave & 1;          // 16 x 2 tile grid
  int i0 = ti * 16, j0 = tj * 16;
  int mn = lane & 15;
  int kb = (lane >> 4) << 1;
  v8f acc = {};
  for (int k = 0; k < DDIM; k += 4) {
    int k0 = k + kb;
    v2f a, b;
    a.x = C[(i0 + mn) * DDIM + k0];
    a.y = C[(i0 + mn) * DDIM + k0 + 1];
    b.x = U[(size_t)k0 * KEIG + (j0 + mn)];
    b.y = U[(size_t)(k0 + 1) * KEIG + (j0 + mn)];
    acc = wmma_f32(a, b, acc);
  }
  int rb = (lane >> 4) << 3;
  for (int r = 0; r < 8; ++r)
    P[(i0 + rb + r) * KEIG + (j0 + mn)] = acc[r];
}

// ---------------------------------------------------------------------------
// 3) prod[256x4096] = P[256x32] @ U^T[32x4096]        (WMMA, 4096 tiles)
// ---------------------------------------------------------------------------
__global__ __launch_bounds__(256) void gemm_PUt_kernel(
    const float* __restrict__ P, const float* __restrict__ U,
    float* __restrict__ prod) {
  int wave = (blockIdx.x * blockDim.x + threadIdx.x) >> 5;
  int lane = threadIdx.x & 31;
  int ti = wave / 256, tj = wave % 256;       // 16 x 256 tile grid
  int i0 = ti * 16, j0 = tj * 16;
  int mn = lane & 15;
  int kb = (lane >> 4) << 1;
  v8f acc = {};
  for (int k = 0; k < KEIG; k += 4) {
    int k0 = k + kb;
    v2f a, b;
    a.x = P[(i0 + mn) * KEIG + k0];
    a.y = P[(i0 + mn) * KEIG + k0 + 1];
    // B[k][n] = U^T[k][n] = U[n][k]
    b.x = U[(size_t)(j0 + mn) * KEIG + k0];
    b.y = U[(size_t)(j0 + mn) * KEIG + k0 + 1];
    acc = wmma_f32(a, b, acc);
  }
  int rb = (lane >> 4) << 3;
  for (int r = 0; r < 8; ++r)
    prod[(size_t)(i0 + rb + r) * DDIM + (j0 + mn)] = acc[r];
}

// ---------------------------------------------------------------------------
// 4) residT = centered - prod (in place into prod), copy into GS work buffer
// ---------------------------------------------------------------------------
__global__ void resid_kernel(const float* __restrict__ centered,
                             float* __restrict__ prod,
                             float* __restrict__ gsV) {
  size_t idx = (size_t)blockIdx.x * blockDim.x + threadIdx.x;
  float r = centered[idx] - prod[idx];
  prod[idx] = r;   // prod now holds residual^T  [B,D]
  gsV[idx]  = r;   // Gram-Schmidt working copy
}

// ---------------------------------------------------------------------------
// 5) Sequential modified Gram-Schmidt + row normalization, one persistent WG.
//    Step t: beta = row t-1; rows i>=t: row_i -= (row_i.beta / beta.beta)*beta
// ---------------------------------------------------------------------------
__global__ __launch_bounds__(1024) void gram_schmidt_kernel(float* __restrict__ V) {
  __shared__ float beta[DDIM];
  __shared__ float red[32];
  __shared__ float bb_s;
  const int tid  = threadIdx.x;
  const int lane = tid & 31;
  const int wid  = tid >> 5;   // 0..31

  for (int t = 1; t < BROW; ++t) {
    // stage beta row in LDS
    for (int idx = tid; idx < DDIM; idx += 1024)
      beta[idx] = V[(size_t)(t - 1) * DDIM + idx];
    __syncthreads();
    // bb = beta.beta (block reduction)
    float p = 0.f;
    for (int idx = tid; idx < DDIM; idx += 1024) p += beta[idx] * beta[idx];
    for (int off = 16; off > 0; off >>= 1) p += __shfl_xor(p, off, 32);
    if (lane == 0) red[wid] = p;
    __syncthreads();
    if (wid == 0) {
      float q = red[lane];
      for (int off = 16; off > 0; off >>= 1) q += __shfl_xor(q, off, 32);
      if (lane == 0) bb_s = q;
    }
    __syncthreads();
    float bb  = bb_s;
    float inv = (bb == 0.f) ? 0.f : 1.f / bb;   // divide_no_nan semantics
    // wave-per-row rank-1 update
    for (int i = t + wid; i < BROW; i += 32) {
      float d = 0.f;
      const float* row = V + (size_t)i * DDIM;
      for (int idx = lane; idx < DDIM; idx += 32) d += row[idx] * beta[idx];
      for (int off = 16; off > 0; off >>= 1) d += __shfl_xor(d, off, 32);
      float coeff = d * inv;
      for (int idx = lane; idx < DDIM; idx += 32)
        V[(size_t)i * DDIM + idx] = row[idx] - coeff * beta[idx];
    }
    __syncthreads();
  }
  // normalize rows
  for (int i = wid; i < BROW; i += 32) {
    float d = 0.f;
    for (int idx = lane; idx < DDIM; idx += 32) {
      float v = V[(size_t)i * DDIM + idx];
      d += v * v;
    }
    for (int off = 16; off > 0; off >>= 1) d += __shfl_xor(d, off, 32);
    float invn = rsqrtf(d);
    for (int idx = lane; idx < DDIM; idx += 32)
      V[(size_t)i * DDIM + idx] *= invn;
  }
}

// ---------------------------------------------------------------------------
// 6) dr[256x256] = orth[256x4096] @ residT^T           (WMMA NT, 256 tiles)
// ---------------------------------------------------------------------------
__global__ __launch_bounds__(256) void gemm_dr_kernel(
    const float* __restrict__ orth, const float* __restrict__ residT,
    float* __restrict__ dr) {
  int wave = (blockIdx.x * blockDim.x + threadIdx.x) >> 5;
  int lane = threadIdx.x & 31;
  int ti = wave >> 4, tj = wave & 15;         // 16 x 16 tile grid
  int i0 = ti * 16, j0 = tj * 16;
  int mn = lane & 15;
  int kb = (lane >> 4) << 1;
  v8f acc = {};
  for (int k = 0; k < DDIM; k += 4) {
    int k0 = k + kb;
    v2f a, b;
    a.x = orth[(size_t)(i0 + mn) * DDIM + k0];
    a.y = orth[(size_t)(i0 + mn) * DDIM + k0 + 1];
    // B[k][n] = residual[k][n] = residT[n][k]
    b.x = residT[(size_t)(j0 + mn) * DDIM + k0];
    b.y = residT[(size_t)(j0 + mn) * DDIM + k0 + 1];
    acc = wmma_f32(a, b, acc);
  }
  int rb = (lane >> 4) << 3;
  for (int r = 0; r < 8; ++r)
    dr[(i0 + rb + r) * BROW + (j0 + mn)] = acc[r];
}

// ---------------------------------------------------------------------------
// 7) Assemble R = [[diag(sigma), ur],[0, dr]]  column-major [288x288]
//    ur[i][b] = sum_d U[d][i] * centered[b][d]
// ---------------------------------------------------------------------------
__global__ void assemble_R_kernel(const float* __restrict__ sigma,
                                  const float* __restrict__ U,
                                  const float* __restrict__ centered,
                                  const float* __restrict__ dr,
                                  float* __restrict__ A) {
  int idx = blockIdx.x * blockDim.x + threadIdx.x;
  if (idx >= NSVD * NSVD) return;
  int i = idx % NSVD;   // row
  int j = idx / NSVD;   // col
  float v;
  if (i < KEIG && j < KEIG) {
    v = (i == j) ? sigma[i] : 0.f;
  } else if (i < KEIG) {
    int b = j - KEIG;
    float acc = 0.f;
    for (int d = 0; d < DDIM; ++d)
      acc += U[(size_t)d * KEIG + i] * centered[(size_t)b * DDIM + d];
    v = acc;
  } else if (j < KEIG) {
    v = 0.f;
  } else {
    v = dr[(i - KEIG) * BROW + (j - KEIG)];
  }
  A[(size_t)j * NSVD + i] = v;
}

// ---------------------------------------------------------------------------
// 8) One-sided Jacobi SVD on A (col-major 288x288). Columns -> u_j * s_j.
//    Round-robin pairing: 144 independent pairs per round, 287 rounds/sweep.
// ---------------------------------------------------------------------------
__global__ __launch_bounds__(160) void jacobi_kernel(float* __restrict__ A) {
  const int p  = threadIdx.x;
  const int NP = NSVD - 1;   // 287
  for (int sweep = 0; sweep < 8; ++sweep) {
    for (int r = 0; r < NP; ++r) {
      if (p < NSVD / 2) {
        int i, j;
        if (p == 0) { i = NSVD - 1; j = r % NP; }
        else        { i = (r + p) % NP; j = (r - p + NP) % NP; }
        float* ci = A + (size_t)i * NSVD;
        float* cj = A + (size_t)j * NSVD;
        float app = 0.f, aqq = 0.f, apq = 0.f;
        for (int x = 0; x < NSVD; ++x) {
          float a = ci[x], b = cj[x];
          app += a * a; aqq += b * b; apq += a * b;
        }
        if (apq * apq > 1e-24f * app * aqq) {
          float tau = (aqq - app) / (2.0f * apq);
          float t   = ((tau >= 0.f) ? 1.f : -1.f) /
                      (fabsf(tau) + sqrtf(1.f + tau * tau));
          float c = rsqrtf(1.f + t * t);
          float s = t * c;
          for (int x = 0; x < NSVD; ++x) {
            float a = ci[x], b = cj[x];
            ci[x] = c * a - s * b;
            cj[x] = s * a + c * b;
          }
        }
      }
      __syncthreads();
    }
  }
}

// ---------------------------------------------------------------------------
// 9) s_j = ||col_j||, sort descending, build Usort (row-major [288][288])
// ---------------------------------------------------------------------------
__global__ __launch_bounds__(NSVD) void svd_finish_kernel(
    const float* __restrict__ A, float* __restrict__ Usort,
    float* __restrict__ s_out) {
  __shared__ float sn[NSVD];
  int j = threadIdx.x;
  const float* cj = A + (size_t)j * NSVD;
  float d = 0.f;
  for (int x = 0; x < NSVD; ++x) d += cj[x] * cj[x];
  float nrm = sqrtf(d);
  sn[j] = nrm;
  __syncthreads();
  int rank = 0;
  for (int i = 0; i < NSVD; ++i) {
    float si = sn[i];
    if (si > nrm || (si == nrm && i < j)) ++rank;
  }
  s_out[rank] = nrm;
  float inv = (nrm > 0.f) ? 1.f / nrm : 0.f;
  for (int x = 0; x < NSVD; ++x)
    Usort[(size_t)x * NSVD + rank] = cj[x] * inv;
}

// ---------------------------------------------------------------------------
// 10) new_eigvec[4096x288] = [eigvec, orth^T] @ Usort   (WMMA, 4608 tiles)
// ---------------------------------------------------------------------------
__global__ __launch_bounds__(256) void gemm_final_kernel(
    const float* __restrict__ eig, const float* __restrict__ orth,
    const float* __restrict__ Us, float* __restrict__ out) {
  int wave = (blockIdx.x * blockDim.x + threadIdx.x) >> 5;
  int lane = threadIdx.x & 31;
  int ti = wave / 18, tj = wave % 18;          // 256 x 18 tile grid
  int i0 = ti * 16, j0 = tj * 16;
  int mn = lane & 15;
  int kb = (lane >> 4) << 1;
  int d  = i0 + mn;
  v8f acc = {};
  for (int k = 0; k < NSVD; k += 4) {
    int k0 = k + kb, k1 = k0 + 1;
    v2f a, b;
    a.x = (k0 < KEIG) ? eig[(size_t)d * KEIG + k0]
                      : orth[(size_t)(k0 - KEIG) * DDIM + d];
    a.y = (k1 < KEIG) ? eig[(size_t)d * KEIG + k1]
                      : orth[(size_t)(k1 - KEIG) * DDIM + d];
    b.x = Us[(size_t)k0 * NSVD + (j0 + mn)];
    b.y = Us[(size_t)k1 * NSVD + (j0 + mn)];
    acc = wmma_f32(a, b, acc);
  }
  int rb = (lane >> 4) << 3;
  for (int r = 0; r < 8; ++r)
    out[(size_t)(i0 + rb + r) * NSVD + (j0 + mn)] = acc[r];
}

// ---------------------------------------------------------------------------
// Host-side launch sequence (all on `stream`, graph-capture safe).
// ---------------------------------------------------------------------------
extern "C" void kernel_launch(void* const* d_in, const int* in_sizes, int n_in,
                              void* d_out, int out_size, void* d_ws,
                              size_t ws_size, hipStream_t stream) {
  (void)in_sizes; (void)n_in; (void)out_size; (void)ws_size;

  const float* X     = (const float*)d_in[0];   // inputs  [256,4096]
  const float* U     = (const float*)d_in[1];   // eigvec  [4096,32]
  const float* sigma = (const float*)d_in[2];   // sigma   [32]
  const float* mean  = (const float*)d_in[3];   // mean    [1,4096]

  float* out      = (float*)d_out;
  float* out_eig  = out;                                  // [4096,288]
  float* out_s    = out + (size_t)DDIM * NSVD;            // [288]
  float* out_mean = out + (size_t)DDIM * NSVD + NSVD;     // [4096]

  // Workspace layout (floats): ~13.6 MB total
  float* ws       = (float*)d_ws;
  float* centered = ws;                                   // 256*4096
  float* prod     = ws + (size_t)BROW * DDIM;             // 256*4096 -> residT
  float* gsV      = ws + (size_t)2 * BROW * DDIM;         // 256*4096 -> orth
  float* P        = ws + (size_t)3 * BROW * DDIM;         // 256*32
  float* dr       = P + BROW * KEIG;                      // 256*256
  float* Rcm      = dr + BROW * BROW;                     // 288*288 col-major
  float* Usort    = Rcm + NSVD * NSVD;                    // 288*288 row-major

  // 1) mean EMA + centering
  mean_center_kernel<<<DDIM / 256, 256, 0, stream>>>(X, mean, out_mean, centered);
  // 2) P = centered @ U        (32 tiles = 4 blocks * 8 waves)
  gemm_P_kernel<<<4, 256, 0, stream>>>(centered, U, P);
  // 3) prod = P @ U^T          (4096 tiles)
  gemm_PUt_kernel<<<512, 256, 0, stream>>>(P, U, prod);
  // 4) residual^T = centered - prod ; copy to GS buffer
  resid_kernel<<<(BROW * DDIM) / 256, 256, 0, stream>>>(centered, prod, gsV);
  // 5) sequential Gram-Schmidt + normalize (persistent single workgroup)
  gram_schmidt_kernel<<<1, 1024, 0, stream>>>(gsV);
  // 6) dr = orth @ residual    (256 tiles)
  gemm_dr_kernel<<<32, 256, 0, stream>>>(gsV, prod, dr);
  // 7) assemble R (col-major) incl. ur = U^T centered^T
  assemble_R_kernel<<<(NSVD * NSVD + 255) / 256, 256, 0, stream>>>(
      sigma, U, centered, dr, Rcm);
  // 8) one-sided Jacobi SVD
  jacobi_kernel<<<1, 160, 0, stream>>>(Rcm);
  // 9) singular values + sorted U factor
  svd_finish_kernel<<<1, NSVD, 0, stream>>>(Rcm, Usort, out_s);
  // 10) new_eigvec = [eigvec, orth^T] @ u   (4608 tiles)
  gemm_final_kernel<<<576, 256, 0, stream>>>(U, gsV, Usort, out_eig);
}